// ST_GAT_31224412242846
// MI455X (gfx1250) — compile-verified
//
#include <hip/hip_runtime.h>
#include <hip/hip_bf16.h>

#define N_NODES 1024
#define BATCH   64
#define IN_CH   12
#define HEADS   8
#define NN      (BATCH * N_NODES)          // 65536 nodes
#define H1      32
#define H2      128
#define K1      6144                        // K/2
#define K2      12288                       // K
#define LRELU   0.2f

typedef __attribute__((ext_vector_type(2))) float v2f;
typedef __attribute__((ext_vector_type(8))) float v8f;

// D = A(16x4) * B(4x16) + C, fp32 WMMA (CDNA5 V_WMMA_F32_16X16X4_F32)
__device__ __forceinline__ v8f wmma4(v2f a, v2f b, v8f c) {
    return __builtin_amdgcn_wmma_f32_16x16x4_f32(
        false, a, false, b, (short)0, c, false, false);
}

__device__ __forceinline__ float sigf(float x) { return 1.f / (1.f + __expf(-x)); }

__device__ __forceinline__ void atomicMaxF(float* a, float v) {
    if (v >= 0.f) atomicMax((int*)a, __float_as_int(v));
    else          atomicMin((unsigned int*)a, __float_as_uint(v));
}

// ---------------- GAT: h = x @ gat_W^T   [65536,12] x [12,96] ----------------
__global__ void k_gat_gemm(const float* __restrict__ x, const float* __restrict__ W,
                           float* __restrict__ h) {
    int wave = (blockIdx.x * blockDim.x + threadIdx.x) >> 5;
    int lane = threadIdx.x & 31, row = lane & 15, hi = lane >> 4;
    int mt = wave / 6, nt = wave % 6;                 // 4096 x 6 tiles
    int m0 = mt * 16, n0 = nt * 16;
    v8f acc = {};
    for (int k0 = 0; k0 < 12; k0 += 4) {
        int ka = k0 + 2 * hi;
        v2f a, b;
        a.x = x[(m0 + row) * 12 + ka];
        a.y = x[(m0 + row) * 12 + ka + 1];
        b.x = W[(n0 + row) * 12 + ka];                // B[k][n] = gat_W[n][k]
        b.y = W[(n0 + row) * 12 + ka + 1];
        acc = wmma4(a, b, acc);
    }
    for (int r = 0; r < 8; ++r)
        h[(m0 + r + 8 * hi) * 96 + n0 + row] = acc[r];
}

// a_src[n,h] = sum_c h[n,h,c]*att_src[h,c]; same for a_dst
__global__ void k_attn_proj(const float* __restrict__ h, const float* __restrict__ as,
                            const float* __restrict__ ad, float* __restrict__ a_src,
                            float* __restrict__ a_dst) {
    int i = blockIdx.x * blockDim.x + threadIdx.x;
    if (i >= NN * HEADS) return;
    int hd = i & 7, n = i >> 3;
    float s1 = 0.f, s2 = 0.f;
    for (int c = 0; c < IN_CH; ++c) {
        float v = h[n * 96 + hd * 12 + c];
        s1 += v * as[hd * 12 + c];
        s2 += v * ad[hd * 12 + c];
    }
    a_src[i] = s1; a_dst[i] = s2;
}

__global__ void k_fill(float* __restrict__ p, float v, int n) {
    int i = blockIdx.x * blockDim.x + threadIdx.x;
    if (i < n) p[i] = v;
}

// ---------------- edge passes (recompute e; L2-resident gathers) ----------------
__global__ void k_edge_max(const int* __restrict__ src, const int* __restrict__ dst,
                           const float* __restrict__ a_src, const float* __restrict__ a_dst,
                           float* __restrict__ m, int E, int Etot) {
    int e = blockIdx.x * blockDim.x + threadIdx.x;
    if (e >= Etot) return;
    int s = (e < E) ? src[e] : (e - E);
    int d = (e < E) ? dst[e] : (e - E);
    for (int hd = 0; hd < HEADS; ++hd) {
        float v = a_src[s * 8 + hd] + a_dst[d * 8 + hd];
        v = v > 0.f ? v : LRELU * v;
        atomicMaxF(&m[d * 8 + hd], v);
    }
}

__global__ void k_edge_sum(const int* __restrict__ src, const int* __restrict__ dst,
                           const float* __restrict__ a_src, const float* __restrict__ a_dst,
                           const float* __restrict__ m, float* __restrict__ den,
                           int E, int Etot) {
    int e = blockIdx.x * blockDim.x + threadIdx.x;
    if (e >= Etot) return;
    int s = (e < E) ? src[e] : (e - E);
    int d = (e < E) ? dst[e] : (e - E);
    for (int hd = 0; hd < HEADS; ++hd) {
        float v = a_src[s * 8 + hd] + a_dst[d * 8 + hd];
        v = v > 0.f ? v : LRELU * v;
        atomicAdd(&den[d * 8 + hd], __expf(v - m[d * 8 + hd]));
    }
}

__global__ void k_edge_agg(const int* __restrict__ src, const int* __restrict__ dst,
                           const float* __restrict__ a_src, const float* __restrict__ a_dst,
                           const float* __restrict__ m, const float* __restrict__ den,
                           const float* __restrict__ h, float* __restrict__ agg,
                           int E, int Etot) {
    int e = blockIdx.x * blockDim.x + threadIdx.x;
    if (e >= Etot) return;
    int s = (e < E) ? src[e] : (e - E);
    int d = (e < E) ? dst[e] : (e - E);
    for (int hd = 0; hd < HEADS; ++hd) {
        float v = a_src[s * 8 + hd] + a_dst[d * 8 + hd];
        v = v > 0.f ? v : LRELU * v;
        float alpha = __expf(v - m[d * 8 + hd]) / (den[d * 8 + hd] + 1e-16f);
        for (int c = 0; c < IN_CH; ++c)
            atomicAdd(&agg[d * 96 + hd * 12 + c], alpha * h[s * 96 + hd * 12 + c]);
    }
}

// g[n,c] = mean_h agg[n,h,c] + bias[c]
__global__ void k_gat_final(const float* __restrict__ agg, const float* __restrict__ bias,
                            float* __restrict__ g) {
    int i = blockIdx.x * blockDim.x + threadIdx.x;
    if (i >= NN * IN_CH) return;
    int c = i % 12, n = i / 12;
    float s = 0.f;
    for (int hd = 0; hd < HEADS; ++hd) s += agg[n * 96 + hd * 12 + c];
    g[i] = s * 0.125f + bias[c];
}

// ---------------- X1 = seq @ Wih1^T   [768,1024] x [1024,128] ----------------
// seq[t][b][node] = g[b*1024+node][t]; row r = t*64+b
__global__ void k_x1_gemm(const float* __restrict__ g, const float* __restrict__ Wih,
                          float* __restrict__ X1) {
    int wave = (blockIdx.x * blockDim.x + threadIdx.x) >> 5;
    int lane = threadIdx.x & 31, row = lane & 15, hi = lane >> 4;
    int mt = wave / 8, nt = wave % 8;                 // 48 x 8 tiles
    int m0 = mt * 16, n0 = nt * 16;
    int r = m0 + row, t = r / 64, b = r % 64;
    v8f acc = {};
    for (int k0 = 0; k0 < N_NODES; k0 += 4) {
        int ka = k0 + 2 * hi;
        v2f a, bv;
        a.x = g[(b * N_NODES + ka) * 12 + t];
        a.y = g[(b * N_NODES + ka + 1) * 12 + t];
        bv.x = Wih[(n0 + row) * N_NODES + ka];
        bv.y = Wih[(n0 + row) * N_NODES + ka + 1];
        acc = wmma4(a, bv, acc);
    }
    for (int rr = 0; rr < 8; ++rr)
        X1[(m0 + rr + 8 * hi) * 128 + n0 + row] = acc[rr];
}

// ---------------- sequential LSTM1+LSTM2: 1 WG, 32 waves, LDS state ----------------
__global__ void __launch_bounds__(1024)
k_lstm_seq(const float* __restrict__ X1,
           const float* __restrict__ Whh1, const float* __restrict__ bih1, const float* __restrict__ bhh1,
           const float* __restrict__ Wih2, const float* __restrict__ Whh2,
           const float* __restrict__ bih2, const float* __restrict__ bhh2,
           float* __restrict__ g1, float* __restrict__ g2, float* __restrict__ h2out) {
    __shared__ float h1s[64 * 32];
    __shared__ float h2s[64 * 128];
    int tid = threadIdx.x;
    int wave = tid >> 5, lane = tid & 31, row = lane & 15, hi = lane >> 4;
    for (int i = tid; i < 64 * 32; i += 1024) h1s[i] = 0.f;
    for (int i = tid; i < 64 * 128; i += 1024) h2s[i] = 0.f;
    float c1r[2] = {0.f, 0.f};
    float c2r[8] = {0.f, 0.f, 0.f, 0.f, 0.f, 0.f, 0.f, 0.f};
    __syncthreads();

    for (int t = 0; t < IN_CH; ++t) {
        // phase 1: g1 = X1[t] + h1 @ Whh1^T + b    (32 tiles == 32 waves)
        {
            int mt = wave >> 3, nt = wave & 7;
            int m0 = mt * 16, n0 = nt * 16;
            v8f acc = {};
            for (int k0 = 0; k0 < H1; k0 += 4) {
                int ka = k0 + 2 * hi;
                v2f a, b;
                a.x = h1s[(m0 + row) * 32 + ka];
                a.y = h1s[(m0 + row) * 32 + ka + 1];
                b.x = Whh1[(n0 + row) * 32 + ka];
                b.y = Whh1[(n0 + row) * 32 + ka + 1];
                acc = wmma4(a, b, acc);
            }
            for (int r = 0; r < 8; ++r) {
                int mm = m0 + r + 8 * hi, nn = n0 + row;
                g1[mm * 128 + nn] = acc[r] + X1[(t * 64 + mm) * 128 + nn] + bih1[nn] + bhh1[nn];
            }
        }
        __threadfence();
        __syncthreads();
        // phase 2: LSTM1 activations (c-state in regs, h-state to LDS)
        for (int q = 0; q < 2; ++q) {
            int p = tid * 2 + q;
            int b = p >> 5, j = p & 31;
            float gi = g1[b * 128 + j];
            float gf = g1[b * 128 + 32 + j];
            float gg = g1[b * 128 + 64 + j];
            float go = g1[b * 128 + 96 + j];
            float c = sigf(gf) * c1r[q] + sigf(gi) * tanhf(gg);
            c1r[q] = c;
            h1s[b * 32 + j] = sigf(go) * tanhf(c);
        }
        __syncthreads();
        // phase 3: g2 = h1 @ Wih2^T + h2 @ Whh2^T + b  (wave owns one 16-col strip, 4 M tiles)
        {
            int n0 = wave * 16;
            for (int mt = 0; mt < 4; ++mt) {
                int m0 = mt * 16;
                v8f acc = {};
                for (int k0 = 0; k0 < H1; k0 += 4) {
                    int ka = k0 + 2 * hi;
                    v2f a, b;
                    a.x = h1s[(m0 + row) * 32 + ka];
                    a.y = h1s[(m0 + row) * 32 + ka + 1];
                    b.x = Wih2[(n0 + row) * 32 + ka];
                    b.y = Wih2[(n0 + row) * 32 + ka + 1];
                    acc = wmma4(a, b, acc);
                }
                for (int k0 = 0; k0 < H2; k0 += 4) {
                    int ka = k0 + 2 * hi;
                    v2f a, b;
                    a.x = h2s[(m0 + row) * 128 + ka];
                    a.y = h2s[(m0 + row) * 128 + ka + 1];
                    b.x = Whh2[(n0 + row) * 128 + ka];
                    b.y = Whh2[(n0 + row) * 128 + ka + 1];
                    acc = wmma4(a, b, acc);
                }
                for (int r = 0; r < 8; ++r) {
                    int mm = m0 + r + 8 * hi, nn = n0 + row;
                    g2[mm * 512 + nn] = acc[r] + bih2[nn] + bhh2[nn];
                }
            }
        }
        __threadfence();
        __syncthreads();
        // phase 4: LSTM2 activations
        for (int q = 0; q < 8; ++q) {
            int p = tid * 8 + q;
            int b = p >> 7, j = p & 127;
            float gi = g2[b * 512 + j];
            float gf = g2[b * 512 + 128 + j];
            float gg = g2[b * 512 + 256 + j];
            float go = g2[b * 512 + 384 + j];
            float c = sigf(gf) * c2r[q] + sigf(gi) * tanhf(gg);
            c2r[q] = c;
            h2s[b * 128 + j] = sigf(go) * tanhf(c);
        }
        __syncthreads();
    }
    for (int i = tid; i < 64 * 128; i += 1024) h2out[i] = h2s[i];
}

// ---------------- o1 = h2_last @ lin1_W^T + b1  [64,128] x [128,6144] ----------------
__global__ void k_head1(const float* __restrict__ hl, const float* __restrict__ W,
                        const float* __restrict__ bias, float* __restrict__ o1) {
    int wave = (blockIdx.x * blockDim.x + threadIdx.x) >> 5;   // 0..383 == nt
    int lane = threadIdx.x & 31, row = lane & 15, hi = lane >> 4;
    int n0 = wave * 16;
    for (int mt = 0; mt < 4; ++mt) {
        int m0 = mt * 16;
        v8f acc = {};
        for (int k0 = 0; k0 < H2; k0 += 4) {
            int ka = k0 + 2 * hi;
            v2f a, b;
            a.x = hl[(m0 + row) * 128 + ka];
            a.y = hl[(m0 + row) * 128 + ka + 1];
            b.x = W[(n0 + row) * 128 + ka];
            b.y = W[(n0 + row) * 128 + ka + 1];
            acc = wmma4(a, b, acc);
        }
        for (int r = 0; r < 8; ++r) {
            int mm = m0 + r + 8 * hi, nn = n0 + row;
            o1[mm * K1 + nn] = acc[r] + bias[nn];
        }
    }
}

// ---------------- out = o1 @ lin2_W^T + b2  [64,6144] x [6144,12288] ----------------
// HBM-bound on lin2_W (302 MB, ~13us floor @ 23.3 TB/s): each wave streams its
// 16-row strip of W exactly once (per-lane sequential b64 + prefetch); the shared
// A matrix (o1) is staged per 64x64 chunk into LDS with the CDNA5 async-DMA path
// (global_load_async_to_lds_b128 + s_wait_asynccnt), then fed to WMMA via ds reads.
// LDS row stride 65 avoids the 16-lanes-same-bank conflict of the A fetch pattern.
#define CK 64
__global__ void __launch_bounds__(128)
k_head2(const float* __restrict__ o1, const float* __restrict__ W,
        const float* __restrict__ bias, float* __restrict__ out) {
    __shared__ float As[64 * 65];
    int tid = threadIdx.x;
    int wave = tid >> 5, lane = tid & 31, row = lane & 15, hi = lane >> 4;
    int n0 = (blockIdx.x * 4 + wave) * 16;            // nt in 0..767
    const float* wrow = &W[(size_t)(n0 + row) * K1];
    v8f acc0 = {}, acc1 = {}, acc2 = {}, acc3 = {};
    for (int k0 = 0; k0 < K1; k0 += CK) {
        // async-stage A chunk [64 rows x CK cols] into LDS (1024 x b128, 8 per thread)
        for (int i = 0; i < 8; ++i) {
            int flat = tid + i * 128;
            int r = flat >> 4;                        // row 0..63
            int kq = (flat & 15) << 2;                // 0..60 step 4
            unsigned lds = (unsigned)(unsigned long long)&As[r * 65 + kq];
            unsigned long long ga =
                (unsigned long long)(const void*)&o1[r * K1 + k0 + kq];
            asm volatile("global_load_async_to_lds_b128 %0, %1, off"
                         :: "v"(lds), "v"(ga) : "memory");
        }
        asm volatile("s_wait_asynccnt 0x0" ::: "memory");
        __syncthreads();
        __builtin_prefetch(&wrow[k0 + CK], 0, 0);
        for (int kk = 0; kk < CK; kk += 4) {
            int ka = kk + 2 * hi;
            v2f b;
            b.x = wrow[k0 + ka];
            b.y = wrow[k0 + ka + 1];
            v2f a0, a1, a2, a3;
            a0.x = As[(row) * 65 + ka];      a0.y = As[(row) * 65 + ka + 1];
            a1.x = As[(16 + row) * 65 + ka]; a1.y = As[(16 + row) * 65 + ka + 1];
            a2.x = As[(32 + row) * 65 + ka]; a2.y = As[(32 + row) * 65 + ka + 1];
            a3.x = As[(48 + row) * 65 + ka]; a3.y = As[(48 + row) * 65 + ka + 1];
            acc0 = wmma4(a0, b, acc0);
            acc1 = wmma4(a1, b, acc1);
            acc2 = wmma4(a2, b, acc2);
            acc3 = wmma4(a3, b, acc3);
        }
        __syncthreads();
    }
    for (int r = 0; r < 8; ++r) {
        int nn = n0 + row;
        float bb = bias[nn];
        out[(r + 8 * hi) * K2 + nn]      = acc0[r] + bb;
        out[(16 + r + 8 * hi) * K2 + nn] = acc1[r] + bb;
        out[(32 + r + 8 * hi) * K2 + nn] = acc2[r] + bb;
        out[(48 + r + 8 * hi) * K2 + nn] = acc3[r] + bb;
    }
}

extern "C" void kernel_launch(void* const* d_in, const int* in_sizes, int n_in,
                              void* d_out, int out_size, void* d_ws, size_t ws_size,
                              hipStream_t stream) {
    const float* x        = (const float*)d_in[0];
    const int*   ei       = (const int*)d_in[1];
    const float* gat_W    = (const float*)d_in[2];
    const float* att_src  = (const float*)d_in[3];
    const float* att_dst  = (const float*)d_in[4];
    const float* gat_bias = (const float*)d_in[5];
    const float* l1_Wih   = (const float*)d_in[6];
    const float* l1_Whh   = (const float*)d_in[7];
    const float* l1_bih   = (const float*)d_in[8];
    const float* l1_bhh   = (const float*)d_in[9];
    const float* l2_Wih   = (const float*)d_in[10];
    const float* l2_Whh   = (const float*)d_in[11];
    const float* l2_bih   = (const float*)d_in[12];
    const float* l2_bhh   = (const float*)d_in[13];
    const float* lin1_W   = (const float*)d_in[14];
    const float* lin1_b   = (const float*)d_in[15];
    const float* lin2_W   = (const float*)d_in[16];
    const float* lin2_b   = (const float*)d_in[17];
    float* out = (float*)d_out;

    const int E = in_sizes[1] / 2;
    const int Etot = E + NN;
    const int* src = ei;
    const int* dst = ei + E;

    // workspace layout (fp32 elements)
    float* ws = (float*)d_ws;
    float* h      = ws;                       // 65536*96
    float* a_src  = h + (size_t)NN * 96;      // 65536*8
    float* a_dst  = a_src + (size_t)NN * 8;
    float* m      = a_dst + (size_t)NN * 8;
    float* den    = m + (size_t)NN * 8;
    float* agg    = den + (size_t)NN * 8;     // 65536*96
    float* g      = agg + (size_t)NN * 96;    // 65536*12
    float* X1     = g + (size_t)NN * 12;      // 768*128
    float* g1     = X1 + 768 * 128;           // 64*128
    float* g2     = g1 + 64 * 128;            // 64*512
    float* h2last = g2 + 64 * 512;            // 64*128
    float* o1     = h2last + 64 * 128;        // 64*6144

    // GAT linear + attention coefficients
    k_gat_gemm<<<(NN / 16) * 6 / 4, 128, 0, stream>>>(x, gat_W, h);
    k_attn_proj<<<(NN * HEADS + 255) / 256, 256, 0, stream>>>(h, att_src, att_dst, a_src, a_dst);

    // init scatter buffers
    k_fill<<<(NN * 8 + 255) / 256, 256, 0, stream>>>(m, -3.402823466e38f, NN * 8);
    k_fill<<<(NN * 8 + 255) / 256, 256, 0, stream>>>(den, 0.f, NN * 8);
    k_fill<<<(NN * 96 + 255) / 256, 256, 0, stream>>>(agg, 0.f, NN * 96);

    // edge softmax + aggregation (3 passes, L2-resident)
    int eb = (Etot + 255) / 256;
    k_edge_max<<<eb, 256, 0, stream>>>(src, dst, a_src, a_dst, m, E, Etot);
    k_edge_sum<<<eb, 256, 0, stream>>>(src, dst, a_src, a_dst, m, den, E, Etot);
    k_edge_agg<<<eb, 256, 0, stream>>>(src, dst, a_src, a_dst, m, den, h, agg, E, Etot);
    k_gat_final<<<(NN * IN_CH + 255) / 256, 256, 0, stream>>>(agg, gat_bias, g);

    // hoisted LSTM1 input GEMM, then sequential recurrence
    k_x1_gemm<<<48 * 8 / 4, 128, 0, stream>>>(g, l1_Wih, X1);
    k_lstm_seq<<<1, 1024, 0, stream>>>(X1, l1_Whh, l1_bih, l1_bhh,
                                       l2_Wih, l2_Whh, l2_bih, l2_bhh,
                                       g1, g2, h2last);

    // MLP head
    k_head1<<<384 / 4, 128, 0, stream>>>(h2last, lin1_W, lin1_b, o1);
    k_head2<<<768 / 4, 128, 0, stream>>>(o1, lin2_W, lin2_b, out);

    (void)n_in; (void)out_size; (void)ws_size;
}